// RingBiasGATv2Layer_15934328668760
// MI455X (gfx1250) — compile-verified
//
#include <hip/hip_runtime.h>
#include <math.h>

typedef __attribute__((ext_vector_type(2))) float v2f;
typedef __attribute__((ext_vector_type(8))) float v8f;

// ---------------------------------------------------------------- utilities

__device__ __forceinline__ void atomic_max_float(float* addr, float val) {
  // classic IEEE monotonic-bits trick; accumulator initialized to -FLT_MAX
  if (val >= 0.0f)
    atomicMax((int*)addr, __float_as_int(val));
  else
    atomicMin((unsigned int*)addr, (unsigned int)__float_as_int(val));
}

// ------------------------------------------------------------ init kernel
__global__ void __launch_bounds__(256)
init_ws_kernel(float* __restrict__ out, float* __restrict__ amax,
               float* __restrict__ den, long long n64, int n4) {
  long long idx = (long long)blockIdx.x * blockDim.x + threadIdx.x;
  if (idx < n64) out[idx] = 0.0f;
  if (idx < n4) {
    amax[idx] = -3.402823466e38f;  // -FLT_MAX
    den[idx]  = 0.0f;
  }
}

// --------------------------------------------------- node GEMMs via WMMA f32
// xl = x@W_l + b_l ; xr = x@W_r + b_r ; xres = x@W_res + b_res
// One wave handles a 16-node tile. A (16x4 f32): lanes 0-15 -> K0,K1 ; lanes
// 16-31 -> K2,K3 (ISA 7.12.2). B (4x16): VGPR0 lanes0-15=K0,l16-31=K2; VGPR1
// K1/K3. D (16x16): VGPR r -> rows r (lanes 0-15) and r+8 (lanes 16-31).
// All 16 B fragments are preloaded into registers so the 32 loads issue as a
// single clause and the 16 WMMAs run back-to-back (no per-WMMA loadcnt-0).
__global__ void __launch_bounds__(128)
node_gemm_kernel(const float* __restrict__ x,
                 const float* __restrict__ Wl, const float* __restrict__ bl,
                 const float* __restrict__ Wr, const float* __restrict__ br,
                 const float* __restrict__ Wres, const float* __restrict__ bres,
                 float* __restrict__ xl, float* __restrict__ xr,
                 float* __restrict__ xres, int ntiles) {
  const int lane = threadIdx.x & 31;
  const int wave = threadIdx.x >> 5;
  const int tile = blockIdx.x * 4 + wave;       // wave-uniform
  if (tile >= ntiles) return;                   // uniform exit, EXEC stays full
  const int row0 = tile * 16;
  const int m    = lane & 15;
  const int sel  = lane >> 4;
  const int koff = sel << 1;                    // 0 or 2

  // Preload A tile: 16 K-chunks of 4 (64 features), shared by all 3 GEMMs.
  const float* xrow = x + (size_t)(row0 + m) * 64;
  v2f a[16];
#pragma unroll
  for (int kc = 0; kc < 16; ++kc)
    a[kc] = *(const v2f*)(xrow + kc * 4 + koff);

  const float* Ws[3]   = {Wl, Wr, Wres};
  const float* bs[3]   = {bl, br, bres};
  float*       outs[3] = {xl, xr, xres};

#pragma unroll
  for (int w = 0; w < 3; ++w) {
    const float* W = Ws[w];
#pragma unroll
    for (int ct = 0; ct < 4; ++ct) {
      const int col = ct * 16 + m;

      // preload the full B K-chain into registers (one load clause)
      v2f b[16];
#pragma unroll
      for (int kc = 0; kc < 16; ++kc) {
        b[kc].x = W[(kc * 4 + koff) * 64 + col];
        b[kc].y = W[(kc * 4 + koff + 1) * 64 + col];
      }

      const float bc = bs[w][col];              // bias broadcasts down rows
      v8f acc;
#pragma unroll
      for (int r = 0; r < 8; ++r) acc[r] = bc;

      // 16 back-to-back WMMAs over K = 64
#pragma unroll
      for (int kc = 0; kc < 16; ++kc)
        acc = __builtin_amdgcn_wmma_f32_16x16x4_f32(
            false, a[kc], false, b[kc], (short)0, acc, false, false);

      float* outp = outs[w];
#pragma unroll
      for (int r = 0; r < 8; ++r) {
        const int row = row0 + r + (sel << 3);
        outp[(size_t)row * 64 + col] = acc[r];
      }
    }
  }
}

// ------------------------------------- fused edge GEMM + attention logits
// One wave handles 16 edges: ee = (edge_attr + hop_bias) @ W_e via WMMA f32
// (K=16 -> 4 chunks), then m = ee + xl[src] + xr[dst], leaky-relu, dot with
// att[h] reduced across the 16-lane half-wave. Col-tile index == head (C=16).
// All W_e fragments (4 heads x 4 chunks) are preloaded once per wave.
__global__ void __launch_bounds__(256)
edge_logits_kernel(const float* __restrict__ edge_attr,
                   const int* __restrict__ edge_index,
                   const int* __restrict__ dist,
                   const float* __restrict__ We,
                   const float* __restrict__ att,
                   const float* __restrict__ bias_per_hop,
                   const float* __restrict__ xl,
                   const float* __restrict__ xr,
                   float* __restrict__ logits, float* __restrict__ amax,
                   int E, int etiles) {
  const int lane = threadIdx.x & 31;
  const int wave = threadIdx.x >> 5;
  const int tile = blockIdx.x * 8 + wave;       // wave-uniform
  if (tile >= etiles) return;
  const int e0   = tile * 16;
  const int m    = lane & 15;
  const int sel  = lane >> 4;
  const int koff = sel << 1;

  // Preload every W_e fragment this lane needs (32 B32 loads, one clause).
  v2f bw[4][4];
#pragma unroll
  for (int h = 0; h < 4; ++h) {
    const int col = h * 16 + m;
#pragma unroll
    for (int kc = 0; kc < 4; ++kc) {
      bw[h][kc].x = We[(kc * 4 + koff) * 64 + col];
      bw[h][kc].y = We[(kc * 4 + koff + 1) * 64 + col];
    }
  }

  // hop bias for A-row m (this lane's matrix row)
  const int eRow = e0 + m;
  const int dRow = edge_index[(size_t)E + eRow];
  int hop = dist[dRow];
  hop = hop < 0 ? 0 : (hop > 3 ? 3 : hop);
  const float hb = bias_per_hop[hop] * 0.1f;

  // A tile: 16 edges x 16 edge-features, biased
  const float* earow = edge_attr + (size_t)eRow * 16;
  v2f a[4];
#pragma unroll
  for (int kc = 0; kc < 4; ++kc) {
    v2f t = *(const v2f*)(earow + kc * 4 + koff);
    t.x += hb;
    t.y += hb;
    a[kc] = t;
  }

  // src/dst for the 8 D-rows this lane owns (rows r and r+8 by half-wave)
  int sArr[8], dArr[8];
#pragma unroll
  for (int r = 0; r < 8; ++r) {
    const int e = e0 + r + (sel << 3);
    sArr[r] = edge_index[e];
    dArr[r] = edge_index[(size_t)E + e];
  }

#pragma unroll
  for (int h = 0; h < 4; ++h) {                 // col tile == head
    const int col = h * 16 + m;
    v8f acc;
#pragma unroll
    for (int r = 0; r < 8; ++r) acc[r] = 0.0f;
#pragma unroll
    for (int kc = 0; kc < 4; ++kc)
      acc = __builtin_amdgcn_wmma_f32_16x16x4_f32(
          false, a[kc], false, bw[h][kc], (short)0, acc, false, false);

    const float av = att[col];                  // att[h][c], c = m
#pragma unroll
    for (int r = 0; r < 8; ++r) {
      float mv = acc[r] + xl[(size_t)sArr[r] * 64 + col]
                        + xr[(size_t)dArr[r] * 64 + col];
      mv = mv > 0.0f ? mv : 0.2f * mv;          // leaky relu
      float contrib = mv * av;
      // sum across the 16 lanes of this half-wave (c dimension)
      contrib += __shfl_xor(contrib, 1, 32);
      contrib += __shfl_xor(contrib, 2, 32);
      contrib += __shfl_xor(contrib, 4, 32);
      contrib += __shfl_xor(contrib, 8, 32);
      if (m == 0) {
        const int e = e0 + r + (sel << 3);
        logits[(size_t)e * 4 + h] = contrib;
        atomic_max_float(&amax[(size_t)dArr[r] * 4 + h], contrib);
      }
    }
  }
}

// ------------------------------------------------ softmax numerator + denom
__global__ void __launch_bounds__(256)
exp_den_kernel(const int* __restrict__ edge_index,
               const float* __restrict__ amax, float* __restrict__ logits,
               float* __restrict__ den, int E) {
  const long long idx = (long long)blockIdx.x * blockDim.x + threadIdx.x;
  if (idx >= (long long)E * 4) return;
  const int e = (int)(idx >> 2);
  const int h = (int)(idx & 3);
  const int d = edge_index[(size_t)E + e];
  const float ex = __expf(logits[idx] - amax[(size_t)d * 4 + h]);
  logits[idx] = ex;                              // reuse buffer for ex
  atomicAdd(&den[(size_t)d * 4 + h], ex);
}

// -------------------------------------------------- weighted scatter-add
__global__ void __launch_bounds__(256)
scatter_kernel(const int* __restrict__ edge_index,
               const float* __restrict__ ex, const float* __restrict__ den,
               const float* __restrict__ xl, float* __restrict__ out, int E) {
  const long long idx = (long long)blockIdx.x * blockDim.x + threadIdx.x;
  if (idx >= (long long)E * 4) return;
  const int e = (int)(idx >> 2);
  const int h = (int)(idx & 3);
  const int s = edge_index[e];
  const int d = edge_index[(size_t)E + e];
  const float alpha = ex[idx] / (den[(size_t)d * 4 + h] + 1e-16f);
  const float* xs = xl + (size_t)s * 64 + h * 16;
  float* od = out + (size_t)d * 64 + h * 16;
#pragma unroll
  for (int c = 0; c < 16; ++c) atomicAdd(&od[c], alpha * xs[c]);
}

// ------------------------------------- bias + LayerNorm + residual (1 wave/node)
__global__ void __launch_bounds__(256)
finalize_kernel(const float* __restrict__ bias, const float* __restrict__ gamma,
                const float* __restrict__ beta, const float* __restrict__ xres,
                float* __restrict__ out, int n) {
  const int lane = threadIdx.x & 31;
  const int wave = threadIdx.x >> 5;
  const int node = blockIdx.x * 8 + wave;
  if (node >= n) return;
  const size_t base = (size_t)node * 64;
  const float v0 = out[base + lane] + bias[lane];
  const float v1 = out[base + lane + 32] + bias[lane + 32];
  float s = v0 + v1;
  s += __shfl_xor(s, 1, 32);  s += __shfl_xor(s, 2, 32);
  s += __shfl_xor(s, 4, 32);  s += __shfl_xor(s, 8, 32);
  s += __shfl_xor(s, 16, 32);
  const float mu = s * (1.0f / 64.0f);
  const float d0 = v0 - mu, d1 = v1 - mu;
  float vs = d0 * d0 + d1 * d1;
  vs += __shfl_xor(vs, 1, 32);  vs += __shfl_xor(vs, 2, 32);
  vs += __shfl_xor(vs, 4, 32);  vs += __shfl_xor(vs, 8, 32);
  vs += __shfl_xor(vs, 16, 32);
  const float inv = rsqrtf(vs * (1.0f / 64.0f) + 1e-5f);
  out[base + lane]      = d0 * inv * gamma[lane]      + beta[lane]      + xres[base + lane];
  out[base + lane + 32] = d1 * inv * gamma[lane + 32] + beta[lane + 32] + xres[base + lane + 32];
}

// ----------------------------------------------------------------- launcher
extern "C" void kernel_launch(void* const* d_in, const int* in_sizes, int n_in,
                              void* d_out, int out_size, void* d_ws,
                              size_t ws_size, hipStream_t stream) {
  const float* x          = (const float*)d_in[0];
  const int*   edge_index = (const int*)d_in[1];
  const float* edge_attr  = (const float*)d_in[2];
  const int*   dist       = (const int*)d_in[3];
  const float* Wl         = (const float*)d_in[4];
  const float* bl         = (const float*)d_in[5];
  const float* Wr         = (const float*)d_in[6];
  const float* br         = (const float*)d_in[7];
  const float* We         = (const float*)d_in[8];
  const float* att        = (const float*)d_in[9];
  const float* bias       = (const float*)d_in[10];
  const float* gamma      = (const float*)d_in[11];
  const float* beta       = (const float*)d_in[12];
  const float* Wres       = (const float*)d_in[13];
  const float* bres       = (const float*)d_in[14];
  const float* bph        = (const float*)d_in[15];

  const int n = in_sizes[0] / 64;    // 100000
  const int E = in_sizes[2] / 16;    // 1600000

  float* ws     = (float*)d_ws;
  float* xl     = ws;
  float* xr     = xl + (size_t)n * 64;
  float* xres   = xr + (size_t)n * 64;
  float* logits = xres + (size_t)n * 64;         // E*4, reused as ex
  float* amax   = logits + (size_t)E * 4;        // n*4
  float* den    = amax + (size_t)n * 4;          // n*4
  float* out    = (float*)d_out;

  const long long n64 = (long long)n * 64;
  init_ws_kernel<<<(int)((n64 + 255) / 256), 256, 0, stream>>>(out, amax, den,
                                                               n64, n * 4);

  const int ntiles = (n + 15) / 16;
  node_gemm_kernel<<<(ntiles + 3) / 4, 128, 0, stream>>>(
      x, Wl, bl, Wr, br, Wres, bres, xl, xr, xres, ntiles);

  const int etiles = (E + 15) / 16;
  edge_logits_kernel<<<(etiles + 7) / 8, 256, 0, stream>>>(
      edge_attr, edge_index, dist, We, att, bph, xl, xr, logits, amax, E,
      etiles);

  const long long eh = (long long)E * 4;
  exp_den_kernel<<<(int)((eh + 255) / 256), 256, 0, stream>>>(edge_index, amax,
                                                              logits, den, E);
  scatter_kernel<<<(int)((eh + 255) / 256), 256, 0, stream>>>(
      edge_index, logits, den, xl, out, E);
  finalize_kernel<<<(n + 7) / 8, 256, 0, stream>>>(bias, gamma, beta, xres,
                                                   out, n);
}